// TemporalConsistencyLoss_22608707846265
// MI455X (gfx1250) — compile-verified
//
#include <hip/hip_runtime.h>

// ---------------------------------------------------------------------------
// TemporalConsistencyLoss on MI455X (gfx1250, wave32).
//
// Memory-bound gather problem: 128 MB of gathered f32 rows -> ~5.5 us at
// 23.3 TB/s HBM. The 3 dot products per pair (vi.vj, vi.vi, vj.vj) are
// computed as diagonals of 16x16 f32 Gram tiles accumulated with
// V_WMMA_F32_16X16X4_F32 (16 pairs per wave). FLOP overhead of the full
// tile is irrelevant under the memory roof; traffic is the optimal 4KB/pair.
// ---------------------------------------------------------------------------

typedef __attribute__((ext_vector_type(2))) float v2f;
typedef __attribute__((ext_vector_type(8))) float v8f;

#define FEAT_D     512
#define NPAIRS     32768
#define PAIRS_PER_WAVE 16
#define NWAVES     (NPAIRS / PAIRS_PER_WAVE)   // 2048
#define WPB        8                            // waves per block
#define TPB        (WPB * 32)                   // 256 threads
#define NBLOCKS    (NWAVES / WPB)               // 256 blocks

__global__ __launch_bounds__(TPB) void tcl_pair_kernel(
    const float* __restrict__ vf,     // [65536, 512]
    const int*   __restrict__ idx1,   // [32768]
    const int*   __restrict__ idx2,   // [32768]
    const float* __restrict__ tg,     // [32768]
    float*       __restrict__ partial_sum,  // [NWAVES]
    float*       __restrict__ partial_cnt)  // [NWAVES]
{
    __shared__ float s_dot[WPB][16];
    __shared__ float s_nii[WPB][16];
    __shared__ float s_njj[WPB][16];

    const int lane   = threadIdx.x & 31;
    const int wv     = threadIdx.x >> 5;
    const int waveId = blockIdx.x * WPB + wv;
    const int pbase  = waveId * PAIRS_PER_WAVE;

    // f32 16x16x4 WMMA operand layout: lane L carries (row/col) = L%16,
    // K offset 0 for lanes 0-15 and 2 for lanes 16-31, two consecutive
    // floats per lane (VGPR0 = K+0, VGPR1 = K+1). Same pattern for A and B
    // when row index == pair index, so one b64 load feeds both roles.
    const int m    = lane & 15;
    const int koff = (lane >> 4) << 1;          // 0 or 2

    const int  p  = pbase + m;
    const long r1 = (long)idx1[p];
    const long r2 = (long)idx2[p];
    const float* ap = vf + r1 * FEAT_D + koff;  // 8B aligned
    const float* bp = vf + r2 * FEAT_D + koff;

    v8f cij = {};   // Vi . Vj^T  -> diag = dot(vi, vj)
    v8f cii = {};   // Vi . Vi^T  -> diag = |vi|^2
    v8f cjj = {};   // Vj . Vj^T  -> diag = |vj|^2

    // 512 / 4 = 128 K-steps, 3 WMMAs each. EXEC is all-1s throughout.
    #pragma unroll 4
    for (int k = 0; k < FEAT_D; k += 4) {
        v2f a = *(const v2f*)(ap + k);
        v2f b = *(const v2f*)(bp + k);
        cij = __builtin_amdgcn_wmma_f32_16x16x4_f32(false, a, false, b,
                                                    (short)0, cij, false, false);
        cii = __builtin_amdgcn_wmma_f32_16x16x4_f32(false, a, false, a,
                                                    (short)0, cii, false, false);
        cjj = __builtin_amdgcn_wmma_f32_16x16x4_f32(false, b, false, b,
                                                    (short)0, cjj, false, false);
    }

    // C/D layout: VGPR r, lanes 0-15 -> (M=r, N=lane); lanes 16-31 ->
    // (M=r+8, N=lane-16). Diagonal element m lives at:
    //   m in [0,8):  VGPR m,   lane m
    //   m in [8,16): VGPR m-8, lane m+16
    #pragma unroll
    for (int r = 0; r < 8; ++r) {
        if (lane == r) {
            s_dot[wv][r] = cij[r]; s_nii[wv][r] = cii[r]; s_njj[wv][r] = cjj[r];
        }
        if (lane == r + 24) {
            s_dot[wv][r + 8] = cij[r]; s_nii[wv][r + 8] = cii[r]; s_njj[wv][r + 8] = cjj[r];
        }
    }
    __syncthreads();

    float val = 0.0f, cnt = 0.0f;
    if (lane < 16) {
        const float dot = s_dot[wv][lane];
        const float nii = s_nii[wv][lane];
        const float njj = s_njj[wv][lane];
        const float den = fmaxf(sqrtf(nii) * sqrtf(njj), 1e-8f);
        const float cosv = dot / den;
        const bool  msk  = tg[pbase + lane] <= 2.0f;
        val = msk ? (1.0f - cosv) : 0.0f;
        cnt = msk ? 1.0f : 0.0f;
    }

    // Fixed-tree wave reduction (lanes >= 16 contribute zeros): deterministic.
    #pragma unroll
    for (int off = 16; off >= 1; off >>= 1) {
        val += __shfl_xor(val, off, 32);
        cnt += __shfl_xor(cnt, off, 32);
    }
    if (lane == 0) {
        partial_sum[waveId] = val;
        partial_cnt[waveId] = cnt;
    }
}

__global__ __launch_bounds__(256) void tcl_reduce_kernel(
    const float* __restrict__ partial_sum,
    const float* __restrict__ partial_cnt,
    const int*   __restrict__ cur_ep,
    const int*   __restrict__ tot_ep,
    float*       __restrict__ out)
{
    __shared__ float s_s[256];
    __shared__ float s_c[256];

    float s = 0.0f, c = 0.0f;
    for (int i = threadIdx.x; i < NWAVES; i += 256) {  // fixed strided order
        s += partial_sum[i];
        c += partial_cnt[i];
    }
    s_s[threadIdx.x] = s;
    s_c[threadIdx.x] = c;
    __syncthreads();

    #pragma unroll
    for (int off = 128; off > 0; off >>= 1) {
        if (threadIdx.x < off) {
            s_s[threadIdx.x] += s_s[threadIdx.x + off];
            s_c[threadIdx.x] += s_c[threadIdx.x + off];
        }
        __syncthreads();
    }

    if (threadIdx.x == 0) {
        int te = *tot_ep;
        if (te < 1) te = 1;
        const float progress = (float)(*cur_ep) / (float)te;
        const float lam = 0.1f + (0.3f - 0.1f) * progress;
        const float total = s_s[0];
        const float num   = s_c[0];
        out[0] = (num > 0.0f) ? (lam * total / fmaxf(num, 1.0f)) : 0.0f;
    }
}

extern "C" void kernel_launch(void* const* d_in, const int* in_sizes, int n_in,
                              void* d_out, int out_size, void* d_ws, size_t ws_size,
                              hipStream_t stream) {
    const float* vf   = (const float*)d_in[0];   // video_features [65536*512]
    const int*   idx1 = (const int*)  d_in[1];   // [32768]
    const int*   idx2 = (const int*)  d_in[2];   // [32768]
    const float* tg   = (const float*)d_in[3];   // [32768]
    const int*   cur  = (const int*)  d_in[4];   // scalar
    const int*   tot  = (const int*)  d_in[5];   // scalar

    float* ps = (float*)d_ws;          // [NWAVES] masked (1-cos) partials
    float* pc = ps + NWAVES;           // [NWAVES] mask counts

    tcl_pair_kernel<<<NBLOCKS, TPB, 0, stream>>>(vf, idx1, idx2, tg, ps, pc);
    tcl_reduce_kernel<<<1, 256, 0, stream>>>(ps, pc, cur, tot, (float*)d_out);
}